// Parser_For_Ovr_6957847019865
// MI455X (gfx1250) — compile-verified
//
#include <hip/hip_runtime.h>
#include <hip/hip_bf16.h>

// ---------------------------------------------------------------------------
// PYSEQM neighbor-list construction for MI455X (gfx1250, wave32).
//
// No matmul in this workload -> no WMMA. The CDNA5 paths exercised instead:
//   * wave32 ballot compaction (__builtin_amdgcn_ballot_w32)
//   * LDS staging + DS atomics + LDS Hillis-Steele scans
//   * bandwidth-optimal single-pass writes of the ~252MB pair output region
//
// Output layout in d_out (all float32, concatenated in reference return order):
//   [0] nmol  [1] molsize
//   [2 .. 2+NMOL)                nHeavy per molecule
//   [2+NMOL .. 2+2*NMOL)         nHydro per molecule
//   offZ = 2+2*NMOL:
//   [offZ + 0*nReal ...)         Z            (nReal)
//   [offZ + 1*nReal ...)         maskd        (nReal)
//   [offZ + 2*nReal ...)         atom_molid   (nReal)
//   offP = offZ + 3*nReal:
//   [offP + 0*nPair ...)         mask         (nPair)
//   [offP + 1*nPair ...)         pair_molid   (nPair)
//   [offP + 2*nPair ...)         ni           (nPair)
//   [offP + 3*nPair ...)         nj           (nPair)
//   [offP + 4*nPair ...)         idxi         (nPair)
//   [offP + 5*nPair ...)         idxj         (nPair)
//   [offP + 6*nPair ...)         xij          (nPair x 3, row-major)
//   [offP + 9*nPair ...)         rij          (nPair)
// ---------------------------------------------------------------------------

#define NMOL     2048
#define MOLSIZE  100
#define CUT2     9.0f            // OUTERCUTOFF^2

// Workspace layout (int32 units)
#define WS_MOLREAL 0             // [NMOL]  per-molecule real-atom count
#define WS_MOLPAIR 2048          // [NMOL]  per-molecule pair count
#define WS_REALOFF 4096          // [NMOL]  exclusive prefix of real counts
#define WS_PAIROFF 6144          // [NMOL]  exclusive prefix of pair counts
#define WS_TOTALS  8192          // [2]     {nReal, nPair}
#define WS_INVREAL 8448          // [NMOL*MOLSIZE]  inv_real_atoms table

static __device__ __forceinline__ unsigned wave_ballot32(bool p) {
#if __has_builtin(__builtin_amdgcn_ballot_w32)
  return __builtin_amdgcn_ballot_w32(p);
#else
  return (unsigned)__ballot(p);
#endif
}

// Shared predicate math (used identically by count & emit kernels so the
// per-molecule pair counts agree bit-exactly with the compaction pass).
static __device__ __forceinline__ float pair_d2(float xi, float yi, float zi,
                                                float xj, float yj, float zj,
                                                float& dx, float& dy, float& dz) {
  dx = xj - xi; dy = yj - yi; dz = zj - zi;       // coord[j] - coord[i]
  return fmaf(dz, dz, fmaf(dy, dy, dx * dx));
}

// ---------------------------------------------------------------------------
// Kernel 1: per-molecule counts (real / heavy / hydro / close pairs)
// ---------------------------------------------------------------------------
__global__ __launch_bounds__(128) void count_kernel(const int* __restrict__ species,
                                                    const float* __restrict__ coords,
                                                    float* __restrict__ out,
                                                    int* __restrict__ ws) {
  const int m   = blockIdx.x;
  const int tid = threadIdx.x;
  __shared__ float sx[MOLSIZE], sy[MOLSIZE], sz[MOLSIZE];
  __shared__ int   sp[MOLSIZE];
  __shared__ int   cnt[4];                        // real, heavy, hydro, pairs

  if (tid < 4) cnt[tid] = 0;
  if (tid < MOLSIZE) {
    sp[tid] = species[m * MOLSIZE + tid];
    const float* c = coords + (size_t)(m * MOLSIZE + tid) * 3;
    sx[tid] = c[0]; sy[tid] = c[1]; sz[tid] = c[2];
  }
  __syncthreads();

  if (tid < MOLSIZE) {                            // DS atomics (wave32, LDS)
    const int s = sp[tid];
    if (s > 0)  atomicAdd(&cnt[0], 1);
    if (s > 1)  atomicAdd(&cnt[1], 1);
    if (s == 1) atomicAdd(&cnt[2], 1);
  }

  int pc = 0;
  for (int idx = tid; idx < MOLSIZE * MOLSIZE; idx += 128) {
    const int i = idx / MOLSIZE, j = idx % MOLSIZE;
    if (i < j && sp[i] > 0 && sp[j] > 0) {
      float dx, dy, dz;
      const float d2 = pair_d2(sx[i], sy[i], sz[i], sx[j], sy[j], sz[j], dx, dy, dz);
      if (d2 < CUT2) ++pc;
    }
  }
  if (pc) atomicAdd(&cnt[3], pc);
  __syncthreads();

  if (tid == 0) {
    ws[WS_MOLREAL + m] = cnt[0];
    ws[WS_MOLPAIR + m] = cnt[3];
    out[2 + m]        = (float)cnt[1];            // nHeavy
    out[2 + NMOL + m] = (float)cnt[2];            // nHydro
  }
}

// ---------------------------------------------------------------------------
// Kernel 2: single-block exclusive scan over 2048 molecules (real & pair)
// ---------------------------------------------------------------------------
__global__ __launch_bounds__(256) void scan_kernel(float* __restrict__ out,
                                                   int* __restrict__ ws) {
  __shared__ int pr[256], pp[256];
  const int t    = threadIdx.x;
  const int base = t * 8;                         // 256 threads x 8 molecules
  int lr[8], lp[8];
  int sr = 0, spc = 0;
  for (int k = 0; k < 8; ++k) {
    lr[k] = ws[WS_MOLREAL + base + k];
    lp[k] = ws[WS_MOLPAIR + base + k];
    sr += lr[k]; spc += lp[k];
  }
  pr[t] = sr; pp[t] = spc;
  __syncthreads();
  for (int off = 1; off < 256; off <<= 1) {       // Hillis-Steele inclusive scan
    const int vr = (t >= off) ? pr[t - off] : 0;
    const int vp = (t >= off) ? pp[t - off] : 0;
    __syncthreads();
    pr[t] += vr; pp[t] += vp;
    __syncthreads();
  }
  int exR = pr[t] - sr, exP = pp[t] - spc;        // exclusive bases
  for (int k = 0; k < 8; ++k) {
    ws[WS_REALOFF + base + k] = exR;
    ws[WS_PAIROFF + base + k] = exP;
    exR += lr[k]; exP += lp[k];
  }
  if (t == 255) { ws[WS_TOTALS + 0] = pr[255]; ws[WS_TOTALS + 1] = pp[255]; }
  if (t == 0)   { out[0] = (float)NMOL; out[1] = (float)MOLSIZE; }
}

// ---------------------------------------------------------------------------
// Kernel 3: per-real-atom outputs (Z, maskd, atom_molid) + inv_real_atoms
// ---------------------------------------------------------------------------
__global__ __launch_bounds__(128) void atom_kernel(const int* __restrict__ species,
                                                   float* __restrict__ out,
                                                   int* __restrict__ ws) {
  const int m   = blockIdx.x;
  const int tid = threadIdx.x;
  __shared__ int sc[128];

  int spv = 0, s = 0;
  if (tid < MOLSIZE) { spv = species[m * MOLSIZE + tid]; s = (spv > 0) ? 1 : 0; }
  sc[tid] = s;
  __syncthreads();
  for (int off = 1; off < 128; off <<= 1) {       // LDS scan of nonblank mask
    const int v = (tid >= off) ? sc[tid - off] : 0;
    __syncthreads();
    sc[tid] += v;
    __syncthreads();
  }
  const int ex       = sc[tid] - s;               // rank within molecule
  const int realBase = ws[WS_REALOFF + m];
  const int nReal    = ws[WS_TOTALS + 0];

  if (tid < MOLSIZE && s) {
    const int g    = realBase + ex;
    const int offZ = 2 + 2 * NMOL;
    ws[WS_INVREAL + m * MOLSIZE + tid] = g;
    out[offZ + g]             = (float)spv;                                   // Z
    out[offZ + nReal + g]     = (float)(tid * (MOLSIZE + 1) + m * MOLSIZE * MOLSIZE); // maskd
    out[offZ + 2 * nReal + g] = (float)m;                                     // atom_molid
  }
}

// ---------------------------------------------------------------------------
// Kernel 4: ordered pair compaction + all per-pair outputs (bandwidth bound)
// ---------------------------------------------------------------------------
__global__ __launch_bounds__(128) void pair_kernel(const int* __restrict__ species,
                                                   const float* __restrict__ coords,
                                                   const float* __restrict__ lcfp,
                                                   float* __restrict__ out,
                                                   int* __restrict__ ws) {
  const int m    = blockIdx.x;
  const int tid  = threadIdx.x;
  const int lane = tid & 31;
  const int wave = tid >> 5;
  __shared__ float sx[MOLSIZE], sy[MOLSIZE], sz[MOLSIZE];
  __shared__ int   sp[MOLSIZE], inv[MOLSIZE];
  __shared__ int   waveCnt[4];
  __shared__ int   sRun;

  if (tid == 0) sRun = 0;
  if (tid < MOLSIZE) {
    sp[tid]  = species[m * MOLSIZE + tid];
    inv[tid] = ws[WS_INVREAL + m * MOLSIZE + tid];
    const float* c = coords + (size_t)(m * MOLSIZE + tid) * 3;
    sx[tid] = c[0]; sy[tid] = c[1]; sz[tid] = c[2];
  }
  const int   nReal   = ws[WS_TOTALS + 0];        // uniform -> scalar loads
  const int   nPair   = ws[WS_TOTALS + 1];
  const int   P       = ws[WS_PAIROFF + m];
  const float lcf     = lcfp[0];
  const int   offP    = 2 + 2 * NMOL + 3 * nReal;
  __syncthreads();

  for (int chunk = 0; chunk < MOLSIZE * MOLSIZE; chunk += 128) {
    const int idx = chunk + tid;
    const int i = idx / MOLSIZE, j = idx % MOLSIZE;
    bool  pred = false;
    float dx = 0.f, dy = 0.f, dz = 0.f, d2 = 0.f;
    if (idx < MOLSIZE * MOLSIZE && i < j && sp[i] > 0 && sp[j] > 0) {
      d2   = pair_d2(sx[i], sy[i], sz[i], sx[j], sy[j], sz[j], dx, dy, dz);
      pred = d2 < CUT2;
    }

    // wave32 ballot + cross-wave LDS prefix => ordered compaction rank
    const unsigned bal = wave_ballot32(pred);
    if (lane == 0) waveCnt[wave] = __popc(bal);
    __syncthreads();
    int base = sRun, total = 0;
    #pragma unroll
    for (int w = 0; w < 4; ++w) { if (w < wave) base += waveCnt[w]; total += waveCnt[w]; }
    const int rank = base + __popc(bal & ((1u << lane) - 1u));
    __syncthreads();
    if (tid == 0) sRun += total;

    if (pred) {
      const int   slot = P + rank;
      const float pd   = d2 * d2;                 // faithful: square, not sqrt
      out[offP + slot]             = (float)((m * MOLSIZE + i) * MOLSIZE + j); // mask
      out[offP + nPair + slot]     = (float)m;                                 // pair_molid
      out[offP + 2 * nPair + slot] = (float)sp[i];                             // ni
      out[offP + 3 * nPair + slot] = (float)sp[j];                             // nj
      out[offP + 4 * nPair + slot] = (float)inv[i];                            // idxi
      out[offP + 5 * nPair + slot] = (float)inv[j];                            // idxj
      float* x3 = out + offP + 6 * nPair + 3 * (size_t)slot;                   // xij
      x3[0] = dx / pd; x3[1] = dy / pd; x3[2] = dz / pd;
      out[offP + 9 * nPair + slot] = pd * lcf;                                 // rij
    }
  }
}

// ---------------------------------------------------------------------------
extern "C" void kernel_launch(void* const* d_in, const int* in_sizes, int n_in,
                              void* d_out, int out_size, void* d_ws, size_t ws_size,
                              hipStream_t stream) {
  const int*   species = (const int*)d_in[0];     // (2048,100) int32
  const float* coords  = (const float*)d_in[1];   // (2048,100,3) f32
  /* d_in[2] = tore (only used for the always-passing parity check) */
  const float* lcf     = (const float*)d_in[3];   // scalar f32
  float* out = (float*)d_out;
  int*   ws  = (int*)d_ws;                        // needs ~852 KB

  count_kernel<<<NMOL, 128, 0, stream>>>(species, coords, out, ws);
  scan_kernel <<<1,    256, 0, stream>>>(out, ws);
  atom_kernel <<<NMOL, 128, 0, stream>>>(species, out, ws);
  pair_kernel <<<NMOL, 128, 0, stream>>>(species, coords, lcf, out, ws);
}